// MORAL_89249420411435
// MI455X (gfx1250) — compile-verified
//
#include <hip/hip_runtime.h>
#include <math.h>

#define F_IN 512
#define HDIM 64
#define C_OUT 7
#define ROWS 64            // rows per block in the WMMA GEMM (4 M-tiles)

typedef __attribute__((ext_vector_type(2))) float v2f;
typedef __attribute__((ext_vector_type(8))) float v8f;

// ---------------------------------------------------------------------------
// gate[k] = (sigmoid(emb[k,:]@embW) >= 0.5) ? 1 : 0   ==  (emb[k,:]@embW >= 0)
// ---------------------------------------------------------------------------
__global__ void k_gate(const float* __restrict__ emb, const float* __restrict__ embW,
                       float* __restrict__ gate) {
    if (blockIdx.x == 0 && threadIdx.x == 0) {
        for (int k = 0; k < 6; ++k) {
            float z = 0.f;
            for (int j = 0; j < HDIM; ++j) z += emb[k * HDIM + j] * embW[j];
            gate[k] = (z >= 0.f) ? 1.f : 0.f;
        }
    }
}

// ---------------------------------------------------------------------------
// degree: deg[i] = 1 (self loop) + #incoming edges at col
// ---------------------------------------------------------------------------
__global__ void k_deg_init(float* __restrict__ deg, int n) {
    int i = blockIdx.x * blockDim.x + threadIdx.x;
    if (i < n) deg[i] = 1.0f;
}

__global__ void k_deg_accum(const long long* __restrict__ col, float* __restrict__ deg, int e) {
    int i = blockIdx.x * blockDim.x + threadIdx.x;
    if (i < e) atomicAdd(&deg[(int)col[i]], 1.0f);
}

__global__ void k_dinv(const float* __restrict__ deg, float* __restrict__ dinv, int n) {
    int i = blockIdx.x * blockDim.x + threadIdx.x;
    if (i < n) {
        float d = deg[i];
        dinv[i] = (d > 0.f) ? rsqrtf(d) : 0.f;
    }
}

// ---------------------------------------------------------------------------
// WMMA fp32 GEMM:  out[N,64] = A[N,512] @ W[512,64]
// Block = 128 threads (4 waves). Each block: 64 rows x 64 cols.
// A panel (64x512 f32 = 128KB LDS) filled with GLOBAL_LOAD_ASYNC_TO_LDS_B128
// (ASYNCcnt), then each wave computes 4 M-tiles of 16x16 so each B fragment
// load feeds 4 v_wmma_f32_16x16x4_f32.
// Fragment layouts per CDNA5 ISA 7.12.2:
//   A 16x4 f32 : lane m=l&15; VGPR0 = K(2h), VGPR1 = K(2h+1), h = l>>4
//   B 4x16 f32 : lane n=l&15; VGPR0 = row K(2h), VGPR1 = row K(2h+1)
//   C/D 16x16  : VGPR r -> M = r + 8h, N = l&15
// ---------------------------------------------------------------------------
__global__ __launch_bounds__(128)
void k_gemm_wmma(const float* __restrict__ A, const float* __restrict__ W,
                 float* __restrict__ out, int n) {
    __shared__ float lds_a[ROWS][F_IN];           // 128 KB
    const int m0  = blockIdx.x * ROWS;
    const int tid = threadIdx.x;

    // ---- async fill of the A panel: 8192 x 16B chunks, 64 per thread ----
    {
        // low 32 bits of the flat shared-memory address == LDS byte address
        unsigned lds_base = (unsigned)(unsigned long long)(&lds_a[0][0]);
#pragma unroll 4
        for (int q = 0; q < 64; ++q) {
            int ci = tid + 128 * q;               // chunk id 0..8191
            int r  = ci >> 7;                     // panel row (128 chunks/row)
            int cc = (ci & 127) << 4;             // byte offset within row
            int row_g = m0 + r;
            if (row_g >= n) row_g = n - 1;        // clamp tail (dup loads OK)
            unsigned goff = (unsigned)row_g * (F_IN * 4) + (unsigned)cc;
            unsigned loff = lds_base + (unsigned)(r * (F_IN * 4) + cc);
            asm volatile("global_load_async_to_lds_b128 %0, %1, %2"
                         :: "v"(loff), "v"(goff), "s"(A) : "memory");
        }
        asm volatile("s_wait_asynccnt 0x0" ::: "memory");
    }
    __syncthreads();

    const int wave = tid >> 5;
    const int lane = tid & 31;
    const int h    = lane >> 4;   // lane half
    const int ln   = lane & 15;   // M index (A) / N index (B,C)
    const int n0   = wave * 16;   // this wave's 16-col slice

    v8f acc[4] = {{}, {}, {}, {}};
#pragma unroll 4
    for (int k0 = 0; k0 < F_IN; k0 += 4) {
        const int ka = k0 + 2 * h;
        v2f b;
        b.x = W[ka       * HDIM + n0 + ln];
        b.y = W[(ka + 1) * HDIM + n0 + ln];
#pragma unroll
        for (int mt = 0; mt < 4; ++mt) {
            v2f a;
            a.x = lds_a[mt * 16 + ln][ka];
            a.y = lds_a[mt * 16 + ln][ka + 1];
            acc[mt] = __builtin_amdgcn_wmma_f32_16x16x4_f32(
                false, a, false, b, (short)0, acc[mt], false, false);
        }
    }

    if (m0 + ROWS <= n) {                          // fast path: no guards
#pragma unroll
        for (int mt = 0; mt < 4; ++mt)
#pragma unroll
            for (int r = 0; r < 8; ++r)
                out[(size_t)(m0 + mt * 16 + r + 8 * h) * HDIM + n0 + ln] = acc[mt][r];
    } else {                                       // tail block only
#pragma unroll
        for (int mt = 0; mt < 4; ++mt)
#pragma unroll
            for (int r = 0; r < 8; ++r) {
                int row = m0 + mt * 16 + r + 8 * h;
                if (row < n) out[(size_t)row * HDIM + n0 + ln] = acc[mt][r];
            }
    }
}

// ---------------------------------------------------------------------------
// xw_s[i,j] = sum_k structure[i,k] * gate[k] * W_s[k,j]   (K = 6, VALU)
// ---------------------------------------------------------------------------
__global__ void k_gemm_s(const float* __restrict__ structure, const float* __restrict__ gate,
                         const float* __restrict__ Ws, float* __restrict__ out, int n) {
    int gid = blockIdx.x * blockDim.x + threadIdx.x;
    int i = gid >> 6, j = gid & 63;
    if (i >= n) return;
    float acc = 0.f;
#pragma unroll
    for (int k = 0; k < 6; ++k)
        acc += structure[(size_t)i * 6 + k] * gate[k] * Ws[k * HDIM + j];
    out[(size_t)i * HDIM + j] = acc;
}

// ---------------------------------------------------------------------------
// xw_r[i,c] = aa2 * (h_f[i,:] @ W_r[0:64,c]) + as2 * (h_s[i,:] @ W_r[64:128,c])
// ---------------------------------------------------------------------------
__global__ void k_gemm_r(const float* __restrict__ hf, const float* __restrict__ hs,
                         const float* __restrict__ Wr,
                         const float* __restrict__ alpha_s, const float* __restrict__ alpha_a,
                         float* __restrict__ out, int n) {
    int gid = blockIdx.x * blockDim.x + threadIdx.x;
    int i = gid >> 3, c = gid & 7;
    if (i >= n || c >= C_OUT) return;
    float as = alpha_s[0], aa = alpha_a[0];
    float denom = fabsf(as) + fabsf(aa);
    float aa2 = 2.f * aa / denom;
    float as2 = 2.f * as / denom;
    const float* hfi = hf + (size_t)i * HDIM;
    const float* hsi = hs + (size_t)i * HDIM;
    float s1 = 0.f, s2 = 0.f;
#pragma unroll 8
    for (int j = 0; j < HDIM; ++j) {
        s1 += hfi[j] * Wr[j * C_OUT + c];
        s2 += hsi[j] * Wr[(HDIM + j) * C_OUT + c];
    }
    out[(size_t)i * C_OUT + c] = aa2 * s1 + as2 * s2;
}

// ---------------------------------------------------------------------------
// self-loop + bias init:  h[i,j] = bias[j] + dinv[i]^2 * xw[i,j]
// (runs before edge scatter; also overwrites d_out poison)
// ---------------------------------------------------------------------------
__global__ void k_selfloop(const float* __restrict__ xw, const float* __restrict__ bias,
                           const float* __restrict__ dinv, float* __restrict__ hout,
                           int n, int hdim) {
    int gid = blockIdx.x * blockDim.x + threadIdx.x;
    if (gid >= n * hdim) return;
    int i = gid / hdim, j = gid - i * hdim;
    float di = dinv[i];
    hout[gid] = bias[j] + di * di * xw[gid];
}

// ---------------------------------------------------------------------------
// edge scatter, H=64: out[col,:] += dinv[row]*dinv[col] * xw[row,:]
// 16 lanes per edge, float4 gather + 4x global_atomic_add_f32
// ---------------------------------------------------------------------------
__global__ void k_scatter64(const long long* __restrict__ row, const long long* __restrict__ col,
                            const float* __restrict__ dinv, const float* __restrict__ xw,
                            float* __restrict__ out, int e) {
    int gid  = blockIdx.x * blockDim.x + threadIdx.x;
    int eidx = gid >> 4;
    if (eidx >= e) return;
    int j = (gid & 15) * 4;
    int r = (int)row[eidx];
    int c = (int)col[eidx];
    float nrm = dinv[r] * dinv[c];
    float4 v = *reinterpret_cast<const float4*>(xw + (size_t)r * HDIM + j);
    float* o = out + (size_t)c * HDIM + j;
    atomicAdd(o + 0, nrm * v.x);
    atomicAdd(o + 1, nrm * v.y);
    atomicAdd(o + 2, nrm * v.z);
    atomicAdd(o + 3, nrm * v.w);
}

// edge scatter, C=7: 8 lanes per edge (one idle)
__global__ void k_scatter7(const long long* __restrict__ row, const long long* __restrict__ col,
                           const float* __restrict__ dinv, const float* __restrict__ xw,
                           float* __restrict__ out, int e) {
    int gid  = blockIdx.x * blockDim.x + threadIdx.x;
    int eidx = gid >> 3;
    int j    = gid & 7;
    if (eidx >= e || j >= C_OUT) return;
    int r = (int)row[eidx];
    int c = (int)col[eidx];
    float nrm = dinv[r] * dinv[c];
    atomicAdd(out + (size_t)c * C_OUT + j, nrm * xw[(size_t)r * C_OUT + j]);
}

// ---------------------------------------------------------------------------
extern "C" void kernel_launch(void* const* d_in, const int* in_sizes, int n_in,
                              void* d_out, int out_size, void* d_ws, size_t ws_size,
                              hipStream_t stream) {
    const float*     feature   = (const float*)d_in[0];
    const float*     structure = (const float*)d_in[1];
    const long long* eidx      = (const long long*)d_in[2];   // int64 [2,E]
    const float*     W_f       = (const float*)d_in[3];
    const float*     b_f       = (const float*)d_in[4];
    const float*     W_s       = (const float*)d_in[5];
    const float*     b_s       = (const float*)d_in[6];
    const float*     W_r       = (const float*)d_in[7];
    const float*     b_r       = (const float*)d_in[8];
    const float*     emb       = (const float*)d_in[9];
    const float*     embW      = (const float*)d_in[10];
    const float*     alpha_s   = (const float*)d_in[11];
    const float*     alpha_a   = (const float*)d_in[12];

    const int n = in_sizes[0] / F_IN;
    const int e = in_sizes[2] / 2;
    const long long* rowi = eidx;
    const long long* coli = eidx + e;

    // workspace (floats): deg[n] | dinv[n] | gate[8] | xw_f[n*64] | xw_s[n*64] | xw_r[n*7]
    float* ws   = (float*)d_ws;
    float* deg  = ws;
    float* dinv = ws + n;
    float* gate = ws + 2 * (size_t)n;
    float* xw_f = gate + 8;
    float* xw_s = xw_f + (size_t)n * HDIM;
    float* xw_r = xw_s + (size_t)n * HDIM;

    // output layout: h_f[n*64] | h_s[n*64] | y[n*7]
    float* h_f = (float*)d_out;
    float* h_s = h_f + (size_t)n * HDIM;
    float* y   = h_s + (size_t)n * HDIM;

    const int B = 256;

    // graph normalization (shared by all three conv layers)
    k_deg_init<<<(n + B - 1) / B, B, 0, stream>>>(deg, n);
    k_deg_accum<<<(e + B - 1) / B, B, 0, stream>>>(coli, deg, e);
    k_dinv<<<(n + B - 1) / B, B, 0, stream>>>(deg, dinv, n);

    // gate
    k_gate<<<1, 32, 0, stream>>>(emb, embW, gate);

    // branch 1: feature GEMM (WMMA + async LDS fill) then GCN aggregate
    k_gemm_wmma<<<(n + ROWS - 1) / ROWS, 128, 0, stream>>>(feature, W_f, xw_f, n);
    k_selfloop<<<((size_t)n * HDIM + B - 1) / B, B, 0, stream>>>(xw_f, b_f, dinv, h_f, n, HDIM);
    k_scatter64<<<((size_t)e * 16 + B - 1) / B, B, 0, stream>>>(rowi, coli, dinv, xw_f, h_f, e);

    // branch 2: gated structure GEMM then GCN aggregate
    k_gemm_s<<<((size_t)n * HDIM + B - 1) / B, B, 0, stream>>>(structure, gate, W_s, xw_s, n);
    k_selfloop<<<((size_t)n * HDIM + B - 1) / B, B, 0, stream>>>(xw_s, b_s, dinv, h_s, n, HDIM);
    k_scatter64<<<((size_t)e * 16 + B - 1) / B, B, 0, stream>>>(rowi, coli, dinv, xw_s, h_s, e);

    // head: fused concat+alpha GEMM then GCN aggregate
    k_gemm_r<<<((size_t)n * 8 + B - 1) / B, B, 0, stream>>>(h_f, h_s, W_r, alpha_s, alpha_a, xw_r, n);
    k_selfloop<<<((size_t)n * C_OUT + B - 1) / B, B, 0, stream>>>(xw_r, b_r, dinv, y, n, C_OUT);
    k_scatter7<<<((size_t)e * 8 + B - 1) / B, B, 0, stream>>>(rowi, coli, dinv, xw_r, y, e);
}